// DP_53755810677168
// MI455X (gfx1250) — compile-verified
//
#include <hip/hip_runtime.h>
#include <hip/hip_bf16.h>

// ---------------------------------------------------------------------------
// DeepMD-style NN potential, fused per-atom kernel for MI455X (gfx1250).
// fp32 end-to-end; GEMMs use V_WMMA_F32_16X16X4_F32 (wave32 WMMA);
// activations use the gfx1250 hardware transcendental V_TANH_F32.
// ---------------------------------------------------------------------------

typedef float v2f __attribute__((ext_vector_type(2)));
typedef float v8f __attribute__((ext_vector_type(8)));

#define BB    4
#define NN    2048
#define MTOT  200          // neighbors per atom
#define MSL   100          // neighbors per type slice
#define E1    25
#define E2    50
#define E3    100
#define M2T   16
#define FH    50
#define ROWS  112          // 100 rows padded to 7 x 16 WMMA M-tiles
#define X1S   28           // K-pad 25->28 (order of 28 mod 64 == 16: conflict-free)
#define X2S   52           // K-pad 50->52 (order of 52 mod 64 == 16: conflict-free)
#define RMINc 0.5f
#define RMAXc 6.0f

// gfx1250 hardware tanh (TRANS32). Prefer the builtin (compiler inserts the
// required TRANS-hazard spacing); fall back to inline asm with a v_nop to
// satisfy the "1 independent op after TRANS before use" rule.
__device__ __forceinline__ float htanh(float x) {
#if __has_builtin(__builtin_amdgcn_tanhf)
  return __builtin_amdgcn_tanhf(x);
#elif __has_builtin(__builtin_amdgcn_tanh_f32)
  return __builtin_amdgcn_tanh_f32(x);
#else
  float y;
  asm volatile("v_tanh_f32 %0, %1\n\tv_nop" : "=v"(y) : "v"(x));
  return y;
#endif
}

struct KParams {
  const float* ImageDR;
  const int*   tmap;
  const float* davg;
  const float* dstd;
  const float* eW0[2][2]; const float* eW1[2][2]; const float* eW2[2][2];
  const float* eb0[2][2]; const float* eb1[2][2]; const float* eb2[2][2];
  const float* fW0[2]; const float* fW1[2]; const float* fW2[2];
  const float* fb0[2]; const float* fb1[2]; const float* fb2[2];
  float* out;
};

// fp32 WMMA fragment layout (ISA 7.12.2):
//  A 16x4 : lanes 0-15 rows M=0..15 hold K={kb,kb+1}; lanes 16-31 hold K={kb+2,kb+3}
//  B 4x16 : assumed mirrored: v.x = B[kb + 2*hi][col], v.y = B[kb+1 + 2*hi][col]
//  C 16x16: VGPR r, lanes 0-15 -> (M=r, N=lane); lanes 16-31 -> (M=r+8, N=lane-16)

__launch_bounds__(128)
__global__ void dp_atom_kernel(KParams P) {
  __shared__ float sRi[MTOT * 4];
  __shared__ float sX1[ROWS * X1S];
  __shared__ float sX2[ROWS * X2S];
  __shared__ float sXYZ[4 * E3];
  __shared__ float sDR[E3 * M2T];
  __shared__ float sFP[2 * FH];
  __shared__ float sH0[FH];
  __shared__ float sH1[FH];

  const int tid  = threadIdx.x;
  const int lane = tid & 31;
  const int wv   = tid >> 5;
  const int lo   = lane & 15;   // column within tile / row within half
  const int hi   = lane >> 4;   // half-wave select
  const int atom = blockIdx.x;
  const int n    = atom % NN;
  const int t    = P.tmap[n];

  // ---------------- Stage P: radial features Ri (normalized) ----------------
  {
    const float* dr = P.ImageDR + (size_t)atom * MTOT * 4;
    const float* da = P.davg + (size_t)t * MTOT * 4;
    const float* ds = P.dstd + (size_t)t * MTOT * 4;
    for (int mi = tid; mi < MTOT; mi += 128) {
      float R  = dr[mi * 4 + 0];
      float dx = dr[mi * 4 + 1], dy = dr[mi * 4 + 2], dz = dr[mi * 4 + 3];
      float safeR = (fabsf(R) > 1e-5f) ? R : 1.0f;
      float u  = (R - RMINc) / (RMAXc - RMINc);
      float sw = u * u * u * (-6.0f * u * u + 15.0f * u - 10.0f) + 1.0f;
      float Srij = 0.0f;
      if (R > 0.0f && R < RMINc)            Srij = 1.0f / safeR;
      else if (R > RMINc && R < RMAXc)      Srij = sw / safeR;
      float inv = (fabsf(R) > 1e-5f) ? (Srij / safeR) : 0.0f;
      float r0 = Srij, r1 = inv * dx, r2 = inv * dy, r3 = inv * dz;
      sRi[mi * 4 + 0] = (r0 - da[mi * 4 + 0]) / ds[mi * 4 + 0];
      sRi[mi * 4 + 1] = (r1 - da[mi * 4 + 1]) / ds[mi * 4 + 1];
      sRi[mi * 4 + 2] = (r2 - da[mi * 4 + 2]) / ds[mi * 4 + 2];
      sRi[mi * 4 + 3] = (r3 - da[mi * 4 + 3]) / ds[mi * 4 + 3];
    }
  }
  for (int i2 = tid; i2 < 4 * E3; i2 += 128) sXYZ[i2] = 0.0f;
  for (int r2 = tid; r2 < ROWS; r2 += 128) {          // zero K-pad cols of X2
    sX2[r2 * X2S + 50] = 0.0f;
    sX2[r2 * X2S + 51] = 0.0f;
  }
  __syncthreads();

  // ---------------- per neighbor-type slice j ----------------
  for (int j = 0; j < 2; ++j) {
    const float* w1  = P.eW0[t][j]; const float* b1v = P.eb0[t][j];
    const float* w2  = P.eW1[t][j]; const float* b2v = P.eb1[t][j];
    const float* w3  = P.eW2[t][j]; const float* b3v = P.eb2[t][j];

    // ---- L1 (1->25): X1[m][c] = tanh(S_m * W[c] + b[c]); zero all padding ----
    for (int idx = tid; idx < ROWS * X1S; idx += 128) {
      int mr = idx / X1S, c = idx % X1S;
      float v = 0.0f;
      if (mr < MSL && c < E1) {
        float s = sRi[(j * MSL + mr) * 4 + 0];
        v = htanh(s * w1[c] + b1v[c]);
      }
      sX1[idx] = v;
    }
    __syncthreads();

    // ---- L2 (25->50, residual [x,x]) via WMMA: each wave owns one 16-col tile ----
    {
      const int Nb  = wv * 16;
      const int col = Nb + lo;
      v2f bfr[7];
#pragma unroll
      for (int ks = 0; ks < 7; ++ks) {
        int k0 = ks * 4 + hi * 2;
        v2f f;
        f.x = (col < E2 && k0     < E1) ? w2[(k0    ) * E2 + col] : 0.0f;
        f.y = (col < E2 && k0 + 1 < E1) ? w2[(k0 + 1) * E2 + col] : 0.0f;
        bfr[ks] = f;
      }
      const float bias = (col < E2) ? b2v[col] : 0.0f;
      const int   c25  = (col >= E1) ? (col - E1) : col;
      for (int mt = 0; mt < 7; ++mt) {
        const int Mb = mt * 16;
        v8f acc = {};
#pragma unroll
        for (int ks = 0; ks < 7; ++ks) {
          int k0 = ks * 4 + hi * 2;
          v2f a;
          a.x = sX1[(Mb + lo) * X1S + k0];
          a.y = sX1[(Mb + lo) * X1S + k0 + 1];
          acc = __builtin_amdgcn_wmma_f32_16x16x4_f32(false, a, false, bfr[ks],
                                                      (short)0, acc, false, false);
        }
        if (col < E2) {
#pragma unroll
          for (int r = 0; r < 8; ++r) {
            int row = Mb + r + 8 * hi;
            // pad rows (>=100) may hold garbage: only used as M-rows in L3, discarded there
            sX2[row * X2S + col] = htanh(acc[r] + bias) + sX1[row * X1S + c25];
          }
        }
      }
    }
    __syncthreads();

    // ---- L3 (50->100, residual) + fused xyz = Ri^T * G (G never materialized) ----
    for (int nt = wv; nt < 7; nt += 4) {       // wave owns cols [16*nt,16*nt+16)
      const int Nb  = nt * 16;
      const int col = Nb + lo;
      v2f bfr[13];
#pragma unroll
      for (int ks = 0; ks < 13; ++ks) {
        int k0 = ks * 4 + hi * 2;
        v2f f;
        f.x = (col < E3 && k0     < E2) ? w3[(k0    ) * E3 + col] : 0.0f;
        f.y = (col < E3 && k0 + 1 < E2) ? w3[(k0 + 1) * E3 + col] : 0.0f;
        bfr[ks] = f;
      }
      const float bias = (col < E3) ? b3v[col] : 0.0f;
      const int   c50  = (col >= E2) ? (col - E2) : col;
      float xp0 = 0.0f, xp1 = 0.0f, xp2 = 0.0f, xp3 = 0.0f;
      for (int mt = 0; mt < 7; ++mt) {
        const int Mb = mt * 16;
        v8f acc = {};
#pragma unroll
        for (int ks = 0; ks < 13; ++ks) {
          int k0 = ks * 4 + hi * 2;
          v2f a;
          a.x = sX2[(Mb + lo) * X2S + k0];
          a.y = sX2[(Mb + lo) * X2S + k0 + 1];
          acc = __builtin_amdgcn_wmma_f32_16x16x4_f32(false, a, false, bfr[ks],
                                                      (short)0, acc, false, false);
        }
        if (col < E3) {
#pragma unroll
          for (int r = 0; r < 8; ++r) {
            int row = Mb + r + 8 * hi;
            if (row < MSL) {
              float g  = htanh(acc[r] + bias) + sX2[row * X2S + c50];
              int   rr = (j * MSL + row) * 4;
              xp0 += sRi[rr + 0] * g;
              xp1 += sRi[rr + 1] * g;
              xp2 += sRi[rr + 2] * g;
              xp3 += sRi[rr + 3] * g;
            }
          }
        }
      }
      // combine the two half-wave row groups (same col in lanes l and l+16)
      xp0 += __shfl_down(xp0, 16, 32);
      xp1 += __shfl_down(xp1, 16, 32);
      xp2 += __shfl_down(xp2, 16, 32);
      xp3 += __shfl_down(xp3, 16, 32);
      if (hi == 0 && col < E3) {               // column owned by exactly one wave
        sXYZ[0 * E3 + col] += xp0;
        sXYZ[1 * E3 + col] += xp1;
        sXYZ[2 * E3 + col] += xp2;
        sXYZ[3 * E3 + col] += xp3;
      }
    }
    __syncthreads();
  }

  // ---------------- DR[e][k] = sum_f (xyz/M)[f][e] * (xyz/M)[f][k] ----------------
  for (int idx = tid; idx < E3 * M2T; idx += 128) {
    int e = idx >> 4, k = idx & 15;
    float s = sXYZ[0 * E3 + e] * sXYZ[0 * E3 + k]
            + sXYZ[1 * E3 + e] * sXYZ[1 * E3 + k]
            + sXYZ[2 * E3 + e] * sXYZ[2 * E3 + k]
            + sXYZ[3 * E3 + e] * sXYZ[3 * E3 + k];
    sDR[idx] = s * (1.0f / (200.0f * 200.0f));
  }
  __syncthreads();

  // ---------------- fit net: 1600 -> 50 (tanh) -> +tanh(50->50) -> 1 ----------------
  if (tid < 100) {
    int g = tid / FH, c = tid % FH;
    const float* w = P.fW0[t] + (size_t)g * 800 * FH;
    const float* d = sDR + g * 800;
    float a = 0.0f;
    for (int q = 0; q < 800; ++q) a += d[q] * w[q * FH + c];
    sFP[g * FH + c] = a;
  }
  __syncthreads();
  if (tid < FH) sH0[tid] = htanh(sFP[tid] + sFP[FH + tid] + P.fb0[t][tid]);
  __syncthreads();
  if (tid < FH) {
    const float* w = P.fW1[t];
    float a = P.fb1[t][tid];
    for (int d2 = 0; d2 < FH; ++d2) a += sH0[d2] * w[d2 * FH + tid];
    sH1[tid] = sH0[tid] + htanh(a);
  }
  __syncthreads();
  if (tid == 0) {
    const float* w = P.fW2[t];
    float a = P.fb2[t][0];
    for (int c = 0; c < FH; ++c) a += sH1[c] * w[c];
    P.out[4 + atom] = a;                      // Ei[b][n], flat (B,N)
  }
}

// Deterministic Etot reduction (no float atomics -> bitwise stable across replays)
__global__ void etot_kernel(float* out) {
  __shared__ float red[256];
  const int b = blockIdx.x;
  const float* Ei = out + 4 + (size_t)b * NN;
  float s = 0.0f;
  for (int i = threadIdx.x; i < NN; i += 256) s += Ei[i];
  red[threadIdx.x] = s;
  __syncthreads();
  for (int st = 128; st > 0; st >>= 1) {
    if (threadIdx.x < st) red[threadIdx.x] += red[threadIdx.x + st];
    __syncthreads();
  }
  if (threadIdx.x == 0) out[b] = red[0];
}

extern "C" void kernel_launch(void* const* d_in, const int* in_sizes, int n_in,
                              void* d_out, int out_size, void* d_ws, size_t ws_size,
                              hipStream_t stream) {
  (void)d_ws; (void)ws_size; (void)out_size;
  KParams P;

  // Top-level flatten order: insertion (list_neigh first) or sorted (ImageDR first).
  bool sortedTop = (n_in > 0 && in_sizes[0] == BB * NN * MTOT * 4);   // 6,553,600
  if (sortedTop) {
    P.ImageDR = (const float*)d_in[0];
    P.tmap    = (const int*)  d_in[1];
    P.davg    = (const float*)d_in[2];
    P.dstd    = (const float*)d_in[3];
    // d_in[4] = list_neigh (unused)
  } else {
    // insertion order: list_neigh, Imagetype_map, ImageDR, davg, dstd
    P.tmap    = (const int*)  d_in[1];
    P.ImageDR = (const float*)d_in[2];
    P.davg    = (const float*)d_in[3];
    P.dstd    = (const float*)d_in[4];
  }

  // params: 'emb' (< 'fit'): [i][j] x {'W':[W0,W1,W2], 'b':[b0,b1,b2]} -> W-list then b-list
  const int embBase = 5;
  for (int i = 0; i < 2; ++i)
    for (int j = 0; j < 2; ++j) {
      int base = embBase + 6 * (2 * i + j);
      P.eW0[i][j] = (const float*)d_in[base + 0];   // 1x25
      P.eW1[i][j] = (const float*)d_in[base + 1];   // 25x50
      P.eW2[i][j] = (const float*)d_in[base + 2];   // 50x100
      P.eb0[i][j] = (const float*)d_in[base + 3];   // 25
      P.eb1[i][j] = (const float*)d_in[base + 4];   // 50
      P.eb2[i][j] = (const float*)d_in[base + 5];   // 100
    }

  // fit dict: sorted keys -> W0,W1,W2,b0,b1,b2 ; insertion -> W0,b0,W1,b1,W2,b2.
  const int fitBase = embBase + 24;
  bool fitSorted = (fitBase + 1 < n_in && in_sizes[fitBase + 1] == FH * FH); // 2500 -> W1
  for (int i = 0; i < 2; ++i) {
    int base = fitBase + 6 * i;
    if (fitSorted) {
      P.fW0[i] = (const float*)d_in[base + 0];
      P.fW1[i] = (const float*)d_in[base + 1];
      P.fW2[i] = (const float*)d_in[base + 2];
      P.fb0[i] = (const float*)d_in[base + 3];
      P.fb1[i] = (const float*)d_in[base + 4];
      P.fb2[i] = (const float*)d_in[base + 5];
    } else {
      P.fW0[i] = (const float*)d_in[base + 0];
      P.fb0[i] = (const float*)d_in[base + 1];
      P.fW1[i] = (const float*)d_in[base + 2];
      P.fb1[i] = (const float*)d_in[base + 3];
      P.fW2[i] = (const float*)d_in[base + 4];
      P.fb2[i] = (const float*)d_in[base + 5];
    }
  }
  P.out = (float*)d_out;

  dp_atom_kernel<<<BB * NN, 128, 0, stream>>>(P);
  etot_kernel<<<BB, 256, 0, stream>>>((float*)d_out);
}